// LinearRef_34986803593457
// MI455X (gfx1250) — compile-verified
//
#include <hip/hip_runtime.h>

typedef __attribute__((ext_vector_type(2))) float v2f;
typedef __attribute__((ext_vector_type(8))) float v8f;

namespace {
constexpr int kNodes   = 50000;
constexpr int kComp    = 16;    // (lmax+1)^2
constexpr int kCin     = 128;
constexpr int kCout    = 128;
constexpr int kWaves   = 8;     // waves per block (wave32)
constexpr int kThreads = kWaves * 32;
constexpr int kKSteps  = kCin / 4;    // 32 K-steps of 4 for f32 WMMA
constexpr int kNTiles  = kCout / 16;  // 8 N-tiles of 16
}

__global__ __launch_bounds__(kThreads)
void sh_block_linear(const float* __restrict__ x,
                     const float* __restrict__ w,
                     const float* __restrict__ pw,
                     float* __restrict__ out) {
  // W_l staged in LDS, pair-interleaved so each B fragment is ONE ds_load_b64:
  //   lw[(c>>1)*2*kCout + 2*d + (c&1)] = W_l[c][d]
  // => (W[c][d], W[c+1][d]) are adjacent for even c.
  __shared__ float lw[kCin * kCout];  // 64 KiB

  const int      l      = blockIdx.y;
  const unsigned deg    = 2u * (unsigned)l + 1u;   // rows per node for this l
  const unsigned base_m = (unsigned)(l * l);
  const int      tiles  = (kNodes * (int)deg) >> 4; // exact multiple, no tail

  const int tid   = threadIdx.x;
  const int wave  = tid >> 5;
  const int lane  = tid & 31;
  const int tile0 = blockIdx.x * kWaves;
  if (tile0 >= tiles) return;  // whole block idle for small l

  // Cooperative swizzled LDS fill: 2x2 (c,d) micro-tile per chunk.
  {
    const float* wl = w + (size_t)l * kCin * kCout;
    #pragma unroll
    for (int it = 0; it < (kCin / 2) * (kCout / 2) / kThreads; ++it) {
      const int i  = tid + it * kThreads;
      const int cp = i >> 6;      // c pair index (kCout/2 == 64)
      const int dp = i & 63;      // d pair index
      const float2 r0 = *(const float2*)(wl + (size_t)(2 * cp) * kCout + 2 * dp);
      const float2 r1 = *(const float2*)(wl + (size_t)(2 * cp + 1) * kCout + 2 * dp);
      *(float4*)(lw + (size_t)cp * (2 * kCout) + 4 * dp) =
          make_float4(r0.x, r1.x, r0.y, r1.y);
    }
  }
  __syncthreads();

  const int tile = tile0 + wave;
  if (tile < tiles) {
    const int half = lane >> 4;   // which half-wave (K pair / M half)
    const int lm   = lane & 15;

    // ---- A-row pointer: A(16x4 f32) lane L holds (M=L%16, K=2*(L/16)+{0,1})
    unsigned r = ((unsigned)tile << 4) + (unsigned)lm;
    unsigned n = r / deg;
    unsigned m = base_m + (r - n * deg);
    const v2f* arow =
        (const v2f*)(x + ((size_t)(n * kComp + m)) * kCin + (size_t)(half * 2));

    // ---- B base (v2f units): element pair (c0=4k, d=16j+lm+64*half)
    //      lives at v2f index 2k*128 + d (d>127: c0+2, d-128)  ->  base + k*256 + j*16
    const v2f* bb = ((const v2f*)lw) + (size_t)(half * (kCout / 2)) + lm;

    v8f acc[kNTiles];
    #pragma unroll
    for (int j = 0; j < kNTiles; ++j) {
      #pragma unroll
      for (int e = 0; e < 8; ++e) acc[j][e] = 0.0f;
    }

    #pragma unroll 4
    for (int k = 0; k < kKSteps; ++k) {
      v2f a = arow[2 * k];  // one global_load_b64 per lane, L0-friendly stride
      #pragma unroll
      for (int j = 0; j < kNTiles; ++j) {
        v2f b = bb[k * 256 + j * 16];  // single ds_load_b64, immediate offset
        acc[j] = __builtin_amdgcn_wmma_f32_16x16x4_f32(
            false, a, false, b, (short)0, acc[j], false, false);
      }
    }

    const float s = pw[l];

    // C/D layout: VGPR v -> (M = half*8 + v, N = 16*j + lm)
    #pragma unroll
    for (int v = 0; v < 8; ++v) {
      unsigned rr = ((unsigned)tile << 4) + (unsigned)(half * 8 + v);
      unsigned nn = rr / deg;
      unsigned mm = base_m + (rr - nn * deg);
      float* orow = out + ((size_t)(nn * kComp + mm)) * kCout + lm;
      #pragma unroll
      for (int j = 0; j < kNTiles; ++j)
        orow[16 * j] = acc[j][v] * s;
    }
  }
}

extern "C" void kernel_launch(void* const* d_in, const int* in_sizes, int n_in,
                              void* d_out, int out_size, void* d_ws, size_t ws_size,
                              hipStream_t stream) {
  const float* x  = (const float*)d_in[0];   // [50000, 16, 128] f32
  const float* w  = (const float*)d_in[1];   // [4, 128, 128] f32
  const float* pw = (const float*)d_in[2];   // [4] f32
  float* out = (float*)d_out;                // [50000, 16, 128] f32

  const int max_tiles = (kNodes * 7) / 16;   // l=3 row-tile count: 21875
  dim3 grid((max_tiles + kWaves - 1) / kWaves, 4);
  sh_block_linear<<<grid, kThreads, 0, stream>>>(x, w, pw, out);
}